// CacheMHA_34952443855341
// MI455X (gfx1250) — compile-verified
//
#include <hip/hip_runtime.h>
#include <math.h>

// ---------------------------------------------------------------------------
// Decode-step MHA with KV cache, MI455X (gfx1250).
// HBM-bound: ~2.15 GB cache-copy traffic -> ~92us floor @ 23.3 TB/s.
// Strategy: fp32 WMMA (v_wmma_f32_16x16x4_f32) for the 4 projections,
// single-pass fused {cache copy + online-softmax attention} for the big part.
// ---------------------------------------------------------------------------

typedef __attribute__((ext_vector_type(2))) float v2f;
typedef __attribute__((ext_vector_type(4))) float v4f;
typedef __attribute__((ext_vector_type(8))) float v8f;

#define D_MODEL  1024
#define N_HEADS  16
#define HEAD_DIM 64
#define BATCH    32
#define KV_LEN   4096
#define S_TOT    4097   // KV_LEN + 1

// D = A(16x4, f32) * B(4x16, f32) + C(16x16, f32)
__device__ __forceinline__ v8f wmma_f32_16x16x4(v2f a, v2f b, v8f c) {
  return __builtin_amdgcn_wmma_f32_16x16x4_f32(
      /*neg_a=*/false, a, /*neg_b=*/false, b,
      /*c_mod=*/(short)0, c, /*reuse_a=*/false, /*reuse_b=*/false);
}

// One 16x16 output tile of Y = X @ W^T, K = D_MODEL.
// A layout: lanes 0-15 hold row m0+hl at K={0,1}; lanes 16-31 at K={2,3}.
// B layout: lanes 0-15 hold W-row n0+hl at K={0,1}; lanes 16-31 at K={2,3}.
__device__ __forceinline__ v8f gemm_tile_xwT(const float* __restrict__ X,
                                             const float* __restrict__ W,
                                             int m0, int n0, int lane) {
  const int hl = lane & 15;
  const int kh = (lane >> 4) << 1;  // 0 or 2
  const float* xrow = X + (size_t)(m0 + hl) * D_MODEL + kh;
  const float* wrow = W + (size_t)(n0 + hl) * D_MODEL + kh;
  v8f c = {};
#pragma unroll 8
  for (int k0 = 0; k0 < D_MODEL; k0 += 4) {
    v2f a = *(const v2f*)(xrow + k0);
    v2f b = *(const v2f*)(wrow + k0);
    c = wmma_f32_16x16x4(a, b, c);
  }
  return c;
}

// ---------------------------------------------------------------------------
// Kernel 1: q = x@Wq^T (to ws), k_new/v_new = x@{Wk,Wv}^T written directly
// into row 4096 of the output caches.  384 waves total.
// ---------------------------------------------------------------------------
__global__ __launch_bounds__(128) void qkv_proj_wmma(
    const float* __restrict__ x,
    const float* __restrict__ Wq, const float* __restrict__ Wk,
    const float* __restrict__ Wv,
    float* __restrict__ q_ws, float* __restrict__ out_k,
    float* __restrict__ out_v) {
  const int gwave = (blockIdx.x * 128 + threadIdx.x) >> 5;  // 0..383
  const int lane  = threadIdx.x & 31;
  const int mat   = gwave >> 7;  // 0=Q 1=K 2=V
  const int tile  = gwave & 127;
  const int m0 = (tile >> 6) << 4;  // 0 or 16 (M=32)
  const int n0 = (tile & 63) << 4;  // N=1024
  const float* W = (mat == 0) ? Wq : (mat == 1) ? Wk : Wv;

  v8f c = gemm_tile_xwT(x, W, m0, n0, lane);

  const int hl = lane & 15;
  const int n  = n0 + hl;
  const int mbase = m0 + ((lane >> 4) << 3);  // VGPR r -> M = mbase + r
  if (mat == 0) {
#pragma unroll
    for (int r = 0; r < 8; ++r)
      q_ws[(size_t)(mbase + r) * D_MODEL + n] = c[r];
  } else {
    float* dst = (mat == 1) ? out_k : out_v;
    const int h = n >> 6, d = n & 63;
#pragma unroll
    for (int r = 0; r < 8; ++r) {
      const int b = mbase + r;
      dst[((size_t)(b * N_HEADS + h) * S_TOT + KV_LEN) * HEAD_DIM + d] = c[r];
    }
  }
}

// ---------------------------------------------------------------------------
// Kernel 2: one workgroup per (b,h).  Single pass: copy cache_k/cache_v rows
// to the output caches AND accumulate online-softmax attention (flash style),
// so K/V are touched exactly once.  16 independent half-wave "units" per
// block, each owning rows s = 16*it + unit with float4-per-lane (B128) I/O.
// ---------------------------------------------------------------------------
__global__ __launch_bounds__(256) void attn_fused(
    const float* __restrict__ cache_k, const float* __restrict__ cache_v,
    const float* __restrict__ q_ws,
    float* __restrict__ out_k, float* __restrict__ out_v,
    float* __restrict__ y_ws) {
  const int bh   = blockIdx.x;  // 0..511  (b*16 + h)
  const int tid  = threadIdx.x;
  const int lane = tid & 31;
  const int wave = tid >> 5;      // 0..7
  const int half = lane >> 4;     // 0/1
  const int hl   = lane & 15;     // lane within half
  const int unit = (wave << 1) | half;  // 0..15
  const int d0   = hl << 2;       // this lane's 4 head-dims

  const size_t src_base = (size_t)bh * (KV_LEN * HEAD_DIM);
  const size_t dst_base = (size_t)bh * (S_TOT * HEAD_DIM);

  const v4f qv = *(const v4f*)(q_ws + (size_t)bh * HEAD_DIM + d0);
  const float scale = 0.125f;  // 1/sqrt(64)

  float m = -3.0e38f, l = 0.0f;
  v4f acc = {};

  for (int it = 0; it < KV_LEN / 16; ++it) {
    const int s = (it << 4) + unit;
    const size_t off = (size_t)s * HEAD_DIM + d0;
    v4f k4 = *(const v4f*)(cache_k + src_base + off);
    v4f v4 = *(const v4f*)(cache_v + src_base + off);
    // stay ~4 iterations (64 rows) ahead of the stream
    __builtin_prefetch(cache_k + src_base + off + 64 * HEAD_DIM, 0, 0);
    __builtin_prefetch(cache_v + src_base + off + 64 * HEAD_DIM, 0, 0);
    *(v4f*)(out_k + dst_base + off) = k4;
    *(v4f*)(out_v + dst_base + off) = v4;

    float p = qv.x * k4.x + qv.y * k4.y + qv.z * k4.z + qv.w * k4.w;
    p += __shfl_xor(p, 1);
    p += __shfl_xor(p, 2);
    p += __shfl_xor(p, 4);
    p += __shfl_xor(p, 8);  // masks <=8 stay inside the 16-lane unit
    const float sc = p * scale;
    const float nm = fmaxf(m, sc);
    const float f  = __expf(m - nm);
    const float w  = __expf(sc - nm);
    acc.x = acc.x * f + w * v4.x;
    acc.y = acc.y * f + w * v4.y;
    acc.z = acc.z * f + w * v4.z;
    acc.w = acc.w * f + w * v4.w;
    l = l * f + w;
    m = nm;
  }

  // Fold in the freshly projected row s = 4096 (written by kernel 1).
  if (unit == 0) {
    const size_t off = (size_t)KV_LEN * HEAD_DIM + d0;
    v4f k4 = *(const v4f*)(out_k + dst_base + off);
    v4f v4 = *(const v4f*)(out_v + dst_base + off);
    float p = qv.x * k4.x + qv.y * k4.y + qv.z * k4.z + qv.w * k4.w;
    p += __shfl_xor(p, 1);
    p += __shfl_xor(p, 2);
    p += __shfl_xor(p, 4);
    p += __shfl_xor(p, 8);
    const float sc = p * scale;
    const float nm = fmaxf(m, sc);
    const float f  = __expf(m - nm);
    const float w  = __expf(sc - nm);
    acc.x = acc.x * f + w * v4.x;
    acc.y = acc.y * f + w * v4.y;
    acc.z = acc.z * f + w * v4.z;
    acc.w = acc.w * f + w * v4.w;
    l = l * f + w;
    m = nm;
  }

  // Combine the 16 partial online-softmax states.
  __shared__ float sm[16];
  __shared__ float sl[16];
  __shared__ __align__(16) float sacc[16][HEAD_DIM];
  if (hl == 0) { sm[unit] = m; sl[unit] = l; }
  *(v4f*)(&sacc[unit][d0]) = acc;
  __syncthreads();

  if (tid < HEAD_DIM) {
    float M = -3.0e38f;
#pragma unroll
    for (int u = 0; u < 16; ++u) M = fmaxf(M, sm[u]);
    float Y = 0.0f, L = 0.0f;
#pragma unroll
    for (int u = 0; u < 16; ++u) {
      const float f = __expf(sm[u] - M);
      Y += f * sacc[u][tid];
      L += f * sl[u];
    }
    y_ws[(size_t)bh * HEAD_DIM + tid] = Y / L;
  }
}

// ---------------------------------------------------------------------------
// Kernel 3: out = y @ Wo^T  (32x1024 @ 1024x1024^T), fp32 WMMA.
// ---------------------------------------------------------------------------
__global__ __launch_bounds__(128) void out_proj_wmma(
    const float* __restrict__ y_ws, const float* __restrict__ Wo,
    float* __restrict__ out) {
  const int gwave = (blockIdx.x * 128 + threadIdx.x) >> 5;  // 0..127
  const int lane  = threadIdx.x & 31;
  const int m0 = (gwave >> 6) << 4;
  const int n0 = (gwave & 63) << 4;

  v8f c = gemm_tile_xwT(y_ws, Wo, m0, n0, lane);

  const int hl = lane & 15;
  const int n  = n0 + hl;
  const int mbase = m0 + ((lane >> 4) << 3);
#pragma unroll
  for (int r = 0; r < 8; ++r)
    out[(size_t)(mbase + r) * D_MODEL + n] = c[r];
}

// ---------------------------------------------------------------------------
extern "C" void kernel_launch(void* const* d_in, const int* in_sizes, int n_in,
                              void* d_out, int out_size, void* d_ws,
                              size_t ws_size, hipStream_t stream) {
  const float* x       = (const float*)d_in[0];  // [32,1,1024]
  const float* cache_k = (const float*)d_in[1];  // [32,16,4096,64]
  const float* cache_v = (const float*)d_in[2];
  const float* Wq      = (const float*)d_in[3];  // [1024,1024]
  const float* Wk      = (const float*)d_in[4];
  const float* Wv      = (const float*)d_in[5];
  const float* Wo      = (const float*)d_in[6];

  float* out   = (float*)d_out;                         // [32,1,1024]
  float* out_k = out + (size_t)BATCH * D_MODEL;         // [32,16,4097,64]
  float* out_v = out_k + (size_t)BATCH * N_HEADS * S_TOT * HEAD_DIM;

  float* q_ws = (float*)d_ws;                           // [32,1024]
  float* y_ws = q_ws + (size_t)BATCH * D_MODEL;         // [32,1024]

  // 1) projections: q -> ws; k_new/v_new -> row 4096 of output caches
  qkv_proj_wmma<<<96, 128, 0, stream>>>(x, Wq, Wk, Wv, q_ws, out_k, out_v);
  // 2) fused cache-copy + online-softmax attention (one block per (b,h))
  attn_fused<<<BATCH * N_HEADS, 256, 0, stream>>>(cache_k, cache_v, q_ws,
                                                  out_k, out_v, y_ws);
  // 3) output projection
  out_proj_wmma<<<32, 128, 0, stream>>>(y_ws, Wo, out);
}